// get_wind_block_spots_41008347742730
// MI455X (gfx1250) — compile-verified
//
#include <hip/hip_runtime.h>

typedef float v2f __attribute__((ext_vector_type(2)));
typedef float v8f __attribute__((ext_vector_type(8)));

// Problem constants (from reference): B=65536, A=14, H=3, D=8, S=512
#define BB 65536
#define AA 14
#define HH 3
#define DD 8
#define SS 512
#define KK (AA * DD)        // 112 = GEMM K
#define RR (BB * HH)        // 196608 = GEMM M (rows)

// Tiling
#define TM 128              // output rows per workgroup
#define TN 64               // spots per workgroup
#define SX 116              // padded LDS stride (floats) for X tile
#define SW 116              // padded LDS stride (floats) for M tile
#define KCH (KK / 4)        // 28 float4 chunks per row

__global__ __launch_bounds__(256)
void wind_spots_wmma_kernel(const float* __restrict__ x,
                            const float* __restrict__ windW,
                            const float* __restrict__ alt,
                            float* __restrict__ out) {
    extern __shared__ float smem[];
    float* Xs = smem;               // TM * SX floats
    float* Ws = smem + TM * SX;     // TN * SW floats

    const int tid  = threadIdx.x;
    const int wave = tid >> 5;
    const int lane = tid & 31;
    const int rowTile = blockIdx.x / (SS / TN);   // 0..1535
    const int colTile = blockIdx.x - rowTile * (SS / TN); // 0..7

    // ---- Stage X tile into LDS: rows r=(b,h), cols k=a*8+d ----
    // global float index of X2[gr][k] = b*336 + a*24 + h*8 + (k&7),  a = k>>3
    #pragma unroll
    for (int i = 0; i < (TM * KCH) / 256; ++i) {          // 14 float4 per thread
        int c  = i * 256 + tid;
        int r  = c / KCH;
        int q  = c - r * KCH;
        int k  = q * 4;
        int a  = k >> 3;
        int dd = k & 7;                                   // 0 or 4
        int gr = rowTile * TM + r;
        int b  = gr / 3;
        int h  = gr - b * 3;
        float4 v = *(const float4*)(x + (size_t)b * (AA * HH * DD)
                                      + (size_t)a * (HH * DD)
                                      + (size_t)h * DD + dd);
        *(float4*)(&Xs[r * SX + k]) = v;
    }

    // ---- Build M tile in LDS on the fly: M[s,k] = K(s, a) * W[s, d] ----
    #pragma unroll
    for (int i = 0; i < (TN * KCH) / 256; ++i) {          // 7 float4 per thread
        int c = i * 256 + tid;
        int r = c / KCH;
        int q = c - r * KCH;
        int k = q * 4;
        int a = k >> 3;
        int d = k & 7;                                    // 0 or 4
        int s = colTile * TN + r;
        float al = alt[s];
        float t1 = fminf(fmaxf((float)a - al + 1.0f, 0.0f), 1.0f);
        float t0 = fminf(fmaxf((float)a - al,        0.0f), 1.0f);
        float kv = t1 - t0;                               // linear-interp weight
        float4 wv = *(const float4*)(windW + s * DD + d);
        float4 mv = make_float4(wv.x * kv, wv.y * kv, wv.z * kv, wv.w * kv);
        *(float4*)(&Ws[r * SW + k]) = mv;
    }

    __syncthreads();

    // ---- GEMM: each wave -> 16 rows x 4 col-subtiles of 16 spots ----
    v8f acc0 = {}, acc1 = {}, acc2 = {}, acc3 = {};
    const int mrow = wave * 16 + (lane & 15);
    const int nrow = lane & 15;
    const int koff = (lane >> 4) << 1;                    // 0 or 2

    #pragma unroll
    for (int kk = 0; kk < KK; kk += 4) {
        v2f a  = *(const v2f*)(&Xs[mrow * SX + kk + koff]);
        v2f b0 = *(const v2f*)(&Ws[( 0 + nrow) * SW + kk + koff]);
        v2f b1 = *(const v2f*)(&Ws[(16 + nrow) * SW + kk + koff]);
        v2f b2 = *(const v2f*)(&Ws[(32 + nrow) * SW + kk + koff]);
        v2f b3 = *(const v2f*)(&Ws[(48 + nrow) * SW + kk + koff]);
        acc0 = __builtin_amdgcn_wmma_f32_16x16x4_f32(false, a, false, b0, (short)0, acc0, false, false);
        acc1 = __builtin_amdgcn_wmma_f32_16x16x4_f32(false, a, false, b1, (short)0, acc1, false, false);
        acc2 = __builtin_amdgcn_wmma_f32_16x16x4_f32(false, a, false, b2, (short)0, acc2, false, false);
        acc3 = __builtin_amdgcn_wmma_f32_16x16x4_f32(false, a, false, b3, (short)0, acc3, false, false);
    }

    // ---- Store C fragments: out[b*S*H + s*H + h], s = colTile*64 + 16n + N ----
    const int nbase = colTile * TN + nrow;
    const int mhalf = (lane >> 4) * 8;                    // VGPR j: M = j (+8 for upper half)
    #pragma unroll
    for (int j = 0; j < 8; ++j) {
        int gr = rowTile * TM + wave * 16 + mhalf + j;
        int b  = gr / 3;
        int h  = gr - b * 3;
        size_t obase = (size_t)b * (SS * HH) + h;
        out[obase + (size_t)(nbase +  0) * HH] = acc0[j];
        out[obase + (size_t)(nbase + 16) * HH] = acc1[j];
        out[obase + (size_t)(nbase + 32) * HH] = acc2[j];
        out[obase + (size_t)(nbase + 48) * HH] = acc3[j];
    }
}

extern "C" void kernel_launch(void* const* d_in, const int* in_sizes, int n_in,
                              void* d_out, int out_size, void* d_ws, size_t ws_size,
                              hipStream_t stream) {
    const float* x     = (const float*)d_in[0];   // (B, A, H, D) fp32
    const float* windW = (const float*)d_in[1];   // (S, D) fp32
    const float* alt   = (const float*)d_in[2];   // (S, 1) fp32
    float* out = (float*)d_out;                   // (B, S, H) fp32
    (void)in_sizes; (void)n_in; (void)out_size; (void)d_ws; (void)ws_size;

    const int grid = (RR / TM) * (SS / TN);       // 1536 * 8 = 12288
    const size_t lds_bytes = (size_t)(TM * SX + TN * SW) * sizeof(float); // ~89 KB
    wind_spots_wmma_kernel<<<grid, 256, lds_bytes, stream>>>(x, windW, alt, out);
}